// EuclideanCodebook_EMA_New_5274219839570
// MI455X (gfx1250) — compile-verified
//
#include <hip/hip_runtime.h>

// EuclideanCodebook EMA update for MI455X (gfx1250), wave32 + WMMA f32.
//
// Shapes: H=1, N=32768 tokens, D=256 dims, C=8192 codes.
// Core: argmin_c( ||e_c||^2 - 2 x.e_c )  -> 137 GFLOP GEMM on f32 matrix cores
// (compute-bound: total HBM traffic ~100MB = ~4us at 23.3 TB/s; the 8MB
// codebook lives in the 192MB L2). Codebook tiles are double-buffered in LDS
// with async VMEM->LDS copies (ASYNCcnt) overlapped with the WMMA chain.

typedef __attribute__((ext_vector_type(2))) float v2f;
typedef __attribute__((ext_vector_type(4))) float v4f;
typedef __attribute__((ext_vector_type(8))) float v8f;

// Payload type for the gfx1250 async VMEM->LDS builtins (from clang proto:
// pointer to 16-byte int vector, AS1 source / AS3 destination).
typedef int v4i_t __attribute__((vector_size(16)));

#define NN 32768
#define DD 256
#define CC 8192
#define DECAY 0.8f
#define OMD 0.2f          // 1 - DECAY
#define EPSF 1e-5f

#define TOK_PER_BLOCK 128   // 8 waves x 16 tokens
#define NWAVES 8
#define CODES_PER_STAGE 32  // 2x 16-code WMMA tiles per stage (2 accumulators)
#define NSTAGES (CC / CODES_PER_STAGE)   // 256
#define KB (DD / 4)         // 64 wmma_f32_16x16x4 steps per K=256
#define ROWF 272            // padded LDS row stride (floats) -> ~conflict-free b64 reads

// CDNA5 async VMEM->LDS path (ASYNCcnt), guarded since not probe-confirmed.
#if defined(__gfx1250__) && __has_builtin(__builtin_amdgcn_global_load_async_to_lds_b128)
#define USE_ASYNC_LDS 1
#else
#define USE_ASYNC_LDS 0
#endif

#define ASYNC_GPTR(p) ((__attribute__((address_space(1))) v4i_t*)(p))
#define ASYNC_LPTR(p) ((__attribute__((address_space(3))) v4i_t*)(p))

__device__ __forceinline__ void wait_asynccnt0() {
#if defined(__gfx1250__)
#if __has_builtin(__builtin_amdgcn_s_wait_asynccnt)
  __builtin_amdgcn_s_wait_asynccnt(0);
#else
  asm volatile("s_wait_asynccnt 0x0" ::: "memory");
#endif
#endif
}

// ---------------------------------------------------------------------------
// Kernel 1: per-code squared norms  y2[c] = sum_d embed[c][d]^2
// ---------------------------------------------------------------------------
__global__ __launch_bounds__(256) void y2_kernel(const float* __restrict__ embed,
                                                 float* __restrict__ y2) {
  const int wave = threadIdx.x >> 5;
  const int lane = threadIdx.x & 31;
  const int c = blockIdx.x * NWAVES + wave;
  const float* row = embed + (size_t)c * DD;
  float s = 0.0f;
#pragma unroll
  for (int j = 0; j < 8; ++j) {
    float v = row[lane + j * 32];
    s += v * v;
  }
#pragma unroll
  for (int mask = 16; mask >= 1; mask >>= 1) s += __shfl_xor(s, mask, 32);
  if (lane == 0) y2[c] = s;
}

// ---------------------------------------------------------------------------
// Kernel 2: assignment GEMM (WMMA f32 16x16x4) with double-buffered LDS
// codebook tiles (async VMEM->LDS), quantize gather, bins + embed_sum via
// hardware f32 atomics. Block: 256 threads = 8 waves, 128 tokens.
// ---------------------------------------------------------------------------
__global__ __launch_bounds__(256) void vq_assign_kernel(
    const float* __restrict__ x, const float* __restrict__ embed,
    const float* __restrict__ y2, float* __restrict__ quantize,
    float* __restrict__ ind_out, float* __restrict__ bins,
    float* embed_sum /* aliases new_embed output segment, pre-zeroed */) {
  __shared__ float etile[2][CODES_PER_STAGE * ROWF];  // 2 x ~34.8 KB (of 320 KB/WGP)
  __shared__ int   sidx[TOK_PER_BLOCK];

  const int tid  = threadIdx.x;
  const int wave = tid >> 5;
  const int lane = tid & 31;
  const int l    = lane & 15;  // column (code) / A-row within half
  const int hi   = lane >> 4;  // which K/M half per WMMA 32-bit layout
  const int waveTok = blockIdx.x * TOK_PER_BLOCK + wave * 16;

  // Per-thread staging geometry: 8 x b128 chunks per thread per 32x256 tile.
  int crow[8], ccol[8];
#pragma unroll
  for (int i = 0; i < 8; ++i) {
    int idx4 = tid + i * 256;   // 2048 float4 per tile
    crow[i] = idx4 >> 6;        // 64 float4 per row
    ccol[i] = (idx4 & 63) * 4;
  }

  // A fragments for this wave's 16 tokens, full K=256, kept in registers and
  // reused across all 256 stages. Per ISA layout: lane(l,hi) vgpr j holds
  // X[m=l][kb*4 + 2*hi + j].
  v2f afrag[KB];
  {
    const float* xrow = x + (size_t)(waveTok + l) * DD + 2 * hi;
#pragma unroll
    for (int kb = 0; kb < KB; ++kb) afrag[kb] = *(const v2f*)(xrow + kb * 4);
  }

  float minv[8];
  int   mini[8];
#pragma unroll
  for (int r = 0; r < 8; ++r) { minv[r] = 3.0e38f; mini[r] = 0; }

  // ---- prologue: stage 0 -> buffer 0 ----
#if USE_ASYNC_LDS
#pragma unroll
  for (int i = 0; i < 8; ++i) {
    __builtin_amdgcn_global_load_async_to_lds_b128(
        ASYNC_GPTR(embed + (size_t)crow[i] * DD + ccol[i]),
        ASYNC_LPTR(&etile[0][crow[i] * ROWF + ccol[i]]),
        0, 0);
  }
  wait_asynccnt0();
#else
#pragma unroll
  for (int i = 0; i < 8; ++i) {
    v4f v = *(const v4f*)(embed + (size_t)crow[i] * DD + ccol[i]);
    *(v4f*)(&etile[0][crow[i] * ROWF + ccol[i]]) = v;
  }
#endif
  __syncthreads();

  for (int stage = 0; stage < NSTAGES; ++stage) {
    const int p = stage & 1;
    const int codeBase = stage * CODES_PER_STAGE;
    const bool haveNext = (stage + 1 < NSTAGES);
    const int nextBase = codeBase + CODES_PER_STAGE;

    // Prefetch next tile into the other buffer; completion is consumed only
    // after the 128-WMMA chain below, so the L2 latency is fully hidden.
#if USE_ASYNC_LDS
    if (haveNext) {
#pragma unroll
      for (int i = 0; i < 8; ++i) {
        __builtin_amdgcn_global_load_async_to_lds_b128(
            ASYNC_GPTR(embed + (size_t)(nextBase + crow[i]) * DD + ccol[i]),
            ASYNC_LPTR(&etile[1 - p][crow[i] * ROWF + ccol[i]]),
            0, 0);
      }
    }
#else
    v4f pref[8];
    if (haveNext) {
#pragma unroll
      for (int i = 0; i < 8; ++i)
        pref[i] = *(const v4f*)(embed + (size_t)(nextBase + crow[i]) * DD + ccol[i]);
    }
#endif

    v8f acc0 = {0.f, 0.f, 0.f, 0.f, 0.f, 0.f, 0.f, 0.f};
    v8f acc1 = {0.f, 0.f, 0.f, 0.f, 0.f, 0.f, 0.f, 0.f};
    const float* b0p = &etile[p][l * ROWF + 2 * hi];          // codes codeBase..+15
    const float* b1p = &etile[p][(16 + l) * ROWF + 2 * hi];   // codes codeBase+16..+31
#pragma unroll
    for (int kb = 0; kb < KB; ++kb) {
      v2f b0 = *(const v2f*)(b0p + kb * 4);
      v2f b1 = *(const v2f*)(b1p + kb * 4);
      acc0 = __builtin_amdgcn_wmma_f32_16x16x4_f32(false, afrag[kb], false, b0,
                                                   (short)0, acc0, false, false);
      acc1 = __builtin_amdgcn_wmma_f32_16x16x4_f32(false, afrag[kb], false, b1,
                                                   (short)0, acc1, false, false);
    }

    // Candidate distances: d = ||e||^2 - 2 x.e  (||x||^2, clip, sqrt monotone)
    const int n0 = codeBase + l;
    const int n1 = codeBase + 16 + l;
    const float y0 = y2[n0];
    const float y1 = y2[n1];
#pragma unroll
    for (int r = 0; r < 8; ++r) {
      float v0 = y0 - 2.0f * acc0[r];          // token m=r+8*hi, code n0
      if (v0 < minv[r]) { minv[r] = v0; mini[r] = n0; }
      float v1 = y1 - 2.0f * acc1[r];
      if (v1 < minv[r]) { minv[r] = v1; mini[r] = n1; }
    }

    // Commit prefetch, then one barrier per stage.
#if USE_ASYNC_LDS
    wait_asynccnt0();
#else
    if (haveNext) {
#pragma unroll
      for (int i = 0; i < 8; ++i)
        *(v4f*)(&etile[1 - p][crow[i] * ROWF + ccol[i]]) = pref[i];
    }
#endif
    __syncthreads();
  }

  // Cross-lane argmin within each 16-lane half (token m = r + 8*hi).
  // Tie-break on smaller code index to match argmax first-occurrence.
#pragma unroll
  for (int r = 0; r < 8; ++r) {
    float v = minv[r];
    int   i = mini[r];
#pragma unroll
    for (int mask = 8; mask >= 1; mask >>= 1) {
      float ov = __shfl_xor(v, mask, 16);
      int   oi = __shfl_xor(i, mask, 16);
      if (ov < v || (ov == v && oi < i)) { v = ov; i = oi; }
    }
    if (l == 0) {
      int m   = r + 8 * hi;
      int tok = waveTok + m;
      ind_out[tok] = (float)i;
      sidx[wave * 16 + m] = i;
      unsafeAtomicAdd(&bins[i], 1.0f);  // global_atomic_add_f32 (no return)
    }
  }
  __syncthreads();

  // Quantize gather + segment-sum scatter; wave handles its 16 tokens,
  // lane covers 8 floats (2 x float4), fully coalesced.
  for (int t = 0; t < 16; ++t) {
    const int tok  = waveTok + t;
    const int code = sidx[wave * 16 + t];
    const float* erow = embed + (size_t)code * DD;
    const float* xrow = x + (size_t)tok * DD;
    float* qrow = quantize + (size_t)tok * DD;
    float* srow = embed_sum + (size_t)code * DD;
#pragma unroll
    for (int j = 0; j < 2; ++j) {
      int col = lane * 4 + j * 128;
      *(v4f*)(qrow + col) = *(const v4f*)(erow + col);
      v4f xv = *(const v4f*)(xrow + col);
      unsafeAtomicAdd(&srow[col + 0], xv.x);
      unsafeAtomicAdd(&srow[col + 1], xv.y);
      unsafeAtomicAdd(&srow[col + 2], xv.z);
      unsafeAtomicAdd(&srow[col + 3], xv.w);
    }
  }
}

// ---------------------------------------------------------------------------
// Kernel 3: new_cluster_size = cs*0.8 + bins*0.2 ; tot = sum(new_cluster_size)
// ---------------------------------------------------------------------------
__global__ __launch_bounds__(256) void finalize_cs_kernel(
    const float* __restrict__ cluster_size, const float* __restrict__ bins,
    float* __restrict__ ncs_out, float* tot) {
  const int c = blockIdx.x * blockDim.x + threadIdx.x;
  float v = cluster_size[c] * DECAY + bins[c] * OMD;
  ncs_out[c] = v;
  __shared__ float red[256];
  red[threadIdx.x] = v;
  __syncthreads();
#pragma unroll
  for (int s = 128; s > 0; s >>= 1) {
    if (threadIdx.x < s) red[threadIdx.x] += red[threadIdx.x + s];
    __syncthreads();
  }
  if (threadIdx.x == 0) unsafeAtomicAdd(tot, red[0]);
}

// ---------------------------------------------------------------------------
// Kernel 4: new_embed_avg = ea*0.8 + embed_sum*0.2 ;
//           new_embed = new_embed_avg / ((ncs+eps)/(tot+C*eps)*tot)
// embed_sum sits in the new_embed output segment (read, then overwritten).
// ---------------------------------------------------------------------------
__global__ __launch_bounds__(256) void finalize_embed_kernel(
    const float* __restrict__ embed_avg, float* new_embed_inout,
    const float* __restrict__ ncs, const float* __restrict__ tot_p,
    float* __restrict__ new_embed_avg) {
  const size_t i = (size_t)blockIdx.x * blockDim.x + threadIdx.x;
  const int c = (int)(i / DD);
  const float esum = new_embed_inout[i];
  const float nea = embed_avg[i] * DECAY + esum * OMD;
  new_embed_avg[i] = nea;
  const float tot = *tot_p;
  const float cs = (ncs[c] + EPSF) / (tot + (float)CC * EPSF) * tot;
  new_embed_inout[i] = nea / cs;
}

// ---------------------------------------------------------------------------
extern "C" void kernel_launch(void* const* d_in, const int* in_sizes, int n_in,
                              void* d_out, int out_size, void* d_ws, size_t ws_size,
                              hipStream_t stream) {
  const float* x            = (const float*)d_in[0];  // [1,N,D]
  const float* embed        = (const float*)d_in[1];  // [1,C,D]
  const float* embed_avg    = (const float*)d_in[2];  // [1,C,D]
  const float* cluster_size = (const float*)d_in[3];  // [1,C]

  // Outputs, flat-concatenated in reference return order.
  float* out       = (float*)d_out;
  float* quantize  = out;                           // N*D
  float* ind       = quantize + (size_t)NN * DD;    // N (indices as float)
  float* new_embed = ind + NN;                      // C*D (doubles as embed_sum)
  float* new_cs    = new_embed + (size_t)CC * DD;   // C
  float* new_ea    = new_cs + CC;                   // C*D

  // Workspace: y2 [C], bins [C], tot [1] -> ~64 KB.
  float* ws   = (float*)d_ws;
  float* y2   = ws;
  float* bins = ws + CC;
  float* tot  = ws + 2 * CC;

  // Zero accumulators every launch (deterministic, graph-capturable memsets).
  (void)hipMemsetAsync(bins, 0, (CC + 64) * sizeof(float), stream);  // bins + tot
  (void)hipMemsetAsync(new_embed, 0, (size_t)CC * DD * sizeof(float), stream);

  y2_kernel<<<CC / NWAVES, 256, 0, stream>>>(embed, y2);
  vq_assign_kernel<<<NN / TOK_PER_BLOCK, 256, 0, stream>>>(
      x, embed, y2, quantize, ind, bins, new_embed);
  finalize_cs_kernel<<<CC / 256, 256, 0, stream>>>(cluster_size, bins, new_cs, tot);
  finalize_embed_kernel<<<(CC * DD) / 256, 256, 0, stream>>>(
      embed_avg, new_embed, new_cs, tot, new_ea);
}